// Attn_12738873000149
// MI455X (gfx1250) — compile-verified
//
#include <hip/hip_runtime.h>
#include <math.h>

typedef __attribute__((ext_vector_type(2))) float v2f;
typedef __attribute__((ext_vector_type(4))) float v4f;
typedef __attribute__((ext_vector_type(8))) float v8f;

#define S_LEN 4096
#define B_LEN 16
#define H_LEN 1024

// ---------------------------------------------------------------------------
// Kernel A: v[b,h] = sum_k hidden[b,k] * W[k,h]   (16x1024 = 16x1024 @ 1024x1024)
// via V_WMMA_F32_16X16X4_F32. One wave per 16-column tile of v.
// Extra block (blockIdx == 64) computes c[b] = hidden[b,:] . b_attn.
// ---------------------------------------------------------------------------
__global__ __launch_bounds__(32) void attn_proj_wmma(
    const float* __restrict__ hidden,   // [B,H]
    const float* __restrict__ W,        // [H(k),H(h)] row-major
    const float* __restrict__ bias,     // [H]
    float* __restrict__ v_out,          // [B,H]
    float* __restrict__ c_out)          // [B]
{
  const int lane = threadIdx.x;     // 0..31, wave32
  if (blockIdx.x == H_LEN / 16) {
    // tiny side computation, off the critical path
    if (lane < B_LEN) {
      float acc = 0.f;
      for (int h = 0; h < H_LEN; ++h)
        acc += hidden[lane * H_LEN + h] * bias[h];
      c_out[lane] = acc;
    }
    return;
  }

  const int n0   = blockIdx.x * 16;   // output column tile
  const int mn   = lane & 15;         // m for A-frag, n for B/C/D frags
  const int half = lane >> 4;         // ISA 16x4 f32 layout: lanes 16-31 hold K=2,3

  v8f acc = {};
  for (int k0 = 0; k0 < H_LEN; k0 += 4) {
    const int k = k0 + half * 2;
    v2f a, b;
    // A (16x4, M x K): lane m = lane&15; VGPR0/1 = K = half*2, half*2+1
    a.x = hidden[mn * H_LEN + k];
    a.y = hidden[mn * H_LEN + k + 1];
    // B (4x16, K x N): lane n = lane&15; VGPR0/1 = rows K = half*2, half*2+1
    b.x = W[(size_t)k       * H_LEN + n0 + mn];
    b.y = W[(size_t)(k + 1) * H_LEN + n0 + mn];
    // 8 args: (neg_a, A, neg_b, B, c_mod, C, reuse_a, reuse_b)
    acc = __builtin_amdgcn_wmma_f32_16x16x4_f32(
        false, a, false, b, (short)0, acc, false, false);
  }

  // C/D layout: VGPR j holds row m = j (lanes 0-15) / m = j+8 (lanes 16-31)
#pragma unroll
  for (int j = 0; j < 8; ++j) {
    const int m = half * 8 + j;
    v_out[m * H_LEN + n0 + mn] = acc[j];
  }
}

// ---------------------------------------------------------------------------
// Kernel B: energies[b,s] = enc[s,b,:] . v[b,:] + c[b]   -> written to d_out
// One wave per (s,b). Streams 256 MB of enc with non-temporal b128 loads.
// ---------------------------------------------------------------------------
__global__ __launch_bounds__(256) void attn_energy(
    const float* __restrict__ enc,   // [S,B,H]
    const float* __restrict__ v,     // [B,H]
    const float* __restrict__ c,     // [B]
    float* __restrict__ energies)    // [B,S]
{
  const int lane = threadIdx.x & 31;
  const int wave = threadIdx.x >> 5;
  const int w    = blockIdx.x * 8 + wave;   // 0 .. S*B-1
  const int s    = w >> 4;
  const int b    = w & (B_LEN - 1);

  const v4f* __restrict__ erow =
      (const v4f*)(enc + ((size_t)s * B_LEN + b) * H_LEN);
  const v4f* __restrict__ vrow = (const v4f*)(v + b * H_LEN);

  float acc = 0.f;
#pragma unroll
  for (int i = 0; i < 8; ++i) {
    v4f e4 = __builtin_nontemporal_load(&erow[i * 32 + lane]); // streamed once
    v4f v4 = vrow[i * 32 + lane];                              // hot in cache
    acc += e4.x * v4.x + e4.y * v4.y + e4.z * v4.z + e4.w * v4.w;
  }
  // wave32 butterfly reduction
#pragma unroll
  for (int off = 16; off > 0; off >>= 1)
    acc += __shfl_xor(acc, off, 32);

  if (lane == 0) energies[(size_t)b * S_LEN + s] = acc + c[b];
}

// ---------------------------------------------------------------------------
// Kernel C: in-place softmax over S for each b. One block per b.
// ---------------------------------------------------------------------------
__global__ __launch_bounds__(256) void attn_softmax(float* __restrict__ out)
{
  __shared__ float red[8];
  const int  b    = blockIdx.x;
  const int  t    = threadIdx.x;
  const int  lane = t & 31;
  const int  wv   = t >> 5;
  float* row = out + (size_t)b * S_LEN;

  float vals[16];
  float m = -INFINITY;
#pragma unroll
  for (int i = 0; i < 16; ++i) {
    vals[i] = row[t + i * 256];
    m = fmaxf(m, vals[i]);
  }
#pragma unroll
  for (int off = 16; off > 0; off >>= 1)
    m = fmaxf(m, __shfl_xor(m, off, 32));
  if (lane == 0) red[wv] = m;
  __syncthreads();
  float bm = red[0];
#pragma unroll
  for (int i = 1; i < 8; ++i) bm = fmaxf(bm, red[i]);
  __syncthreads();

  float sum = 0.f;
#pragma unroll
  for (int i = 0; i < 16; ++i) {
    vals[i] = __expf(vals[i] - bm);
    sum += vals[i];
  }
#pragma unroll
  for (int off = 16; off > 0; off >>= 1)
    sum += __shfl_xor(sum, off, 32);
  if (lane == 0) red[wv] = sum;
  __syncthreads();
  float bs = 0.f;
#pragma unroll
  for (int i = 0; i < 8; ++i) bs += red[i];
  const float inv = 1.0f / bs;
#pragma unroll
  for (int i = 0; i < 16; ++i) row[t + i * 256] = vals[i] * inv;
}

// ---------------------------------------------------------------------------
extern "C" void kernel_launch(void* const* d_in, const int* in_sizes, int n_in,
                              void* d_out, int out_size, void* d_ws, size_t ws_size,
                              hipStream_t stream) {
  const float* hidden = (const float*)d_in[0];   // [1,B,H]
  const float* enc    = (const float*)d_in[1];   // [S,B,H]
  const float* W      = (const float*)d_in[2];   // [H,H]
  const float* bias   = (const float*)d_in[3];   // [H]
  float* out = (float*)d_out;                    // [B,1,S]

  float* ws_v = (float*)d_ws;                    // B*H floats
  float* ws_c = ws_v + B_LEN * H_LEN;            // B floats

  attn_proj_wmma<<<H_LEN / 16 + 1, 32, 0, stream>>>(hidden, W, bias, ws_v, ws_c);
  attn_energy<<<(S_LEN * B_LEN) / 8, 256, 0, stream>>>(enc, ws_v, ws_c, out);
  attn_softmax<<<B_LEN, 256, 0, stream>>>(out);
}